// QLSTM_65481071404450
// MI455X (gfx1250) — compile-verified
//
#include <hip/hip_runtime.h>

typedef float v2f __attribute__((ext_vector_type(2)));
typedef float v4f __attribute__((ext_vector_type(4)));
typedef float v8f __attribute__((ext_vector_type(8)));

#define T_STEPS 1024
#define BATCH   512
#define DIM     128
#define NH      8
#define FAN     (DIM + NH)   // 136
#define ZSTRIDE 16           // padded width of Zx rows
#define TILES_PER_WAVE 4

// ---------------------------------------------------------------------------
// Packed weight matrix Wcat [128 x 16]:
//   cols 0..7  = Wu[h, d]   (g pre-activations)
//   col  8     = Wf[0, d]   (f gate uses only row 0 of z)
//   col  9     = Wi[0, d]
//   col 10     = Wo[0, d]
//   cols 11..15 = 0 (pad)
// ---------------------------------------------------------------------------
__device__ __forceinline__ float wcat(const float* __restrict__ Wf,
                                      const float* __restrict__ Wi,
                                      const float* __restrict__ Wu,
                                      const float* __restrict__ Wo,
                                      int d, int h) {
    if (h < 8)   return Wu[h * FAN + d];
    if (h == 8)  return Wf[d];
    if (h == 9)  return Wi[d];
    if (h == 10) return Wo[d];
    return 0.0f;
}

// ---------------------------------------------------------------------------
// Phase 1: Zx[T*B, 16] = x[T*B, 128] @ Wcat[128, 16] + bias   (WMMA f32)
// One wave handles 4 consecutive 16-row tiles; B matrix pinned in 64 VGPRs.
// Per tile: 32 outstanding global_load_b64 -> 32 chained WMMAs (no per-op
// full-drain s_wait_loadcnt stalls).
// ---------------------------------------------------------------------------
__global__ __launch_bounds__(256) void qlstm_gemm_kernel(
    const float* __restrict__ x,
    const float* __restrict__ Wf, const float* __restrict__ bf,
    const float* __restrict__ Wi, const float* __restrict__ bi,
    const float* __restrict__ Wu, const float* __restrict__ bu,
    const float* __restrict__ Wo, const float* __restrict__ bo,
    float* __restrict__ zx)
{
    const int lane  = threadIdx.x & 31;
    const int wave  = threadIdx.x >> 5;
    const int tile0 = (blockIdx.x * 8 + wave) * TILES_PER_WAVE;

    const int col   = lane & 15;          // N column owned in B/C layout
    const int khalf = (lane >> 4) << 1;   // 0 (lanes 0-15) or 2 (lanes 16-31)

    // Preload B: VGPR0 of step kk holds B[K=4kk+khalf][col], VGPR1 -> K+1
    v2f breg[32];
#pragma unroll
    for (int kk = 0; kk < 32; ++kk) {
        const int d = kk * 4 + khalf;
        breg[kk].x = wcat(Wf, Wi, Wu, Wo, d,     col);
        breg[kk].y = wcat(Wf, Wi, Wu, Wo, d + 1, col);
    }

    const float bias = (col < 8)  ? bu[col]
                     : (col == 8) ? bf[0]
                     : (col == 9) ? bi[0]
                     : (col == 10)? bo[0] : 0.0f;

    for (int rep = 0; rep < TILES_PER_WAVE; ++rep) {
        const int row0 = (tile0 + rep) * 16;
        // A layout: lane L<16 -> (M=row0+L, K=k0..k0+1), lanes 16-31 K+=2
        const float* aptr = x + (size_t)(row0 + (lane & 15)) * DIM + khalf;

        // Stage 1: issue all 32 b64 loads into distinct VGPRs (pipelined)
        v2f afrag[32];
#pragma unroll
        for (int kk = 0; kk < 32; ++kk)
            afrag[kk] = *(const v2f*)(aptr + kk * 4);

        // Stage 2: drain into the WMMA accumulation chain
        v8f acc = {};
#pragma unroll
        for (int kk = 0; kk < 32; ++kk)
            acc = __builtin_amdgcn_wmma_f32_16x16x4_f32(
                false, afrag[kk], false, breg[kk],
                (short)0, acc, false, false);

        // C/D layout: VGPR v -> (M=v, N=lane) lanes 0-15, (M=v+8) lanes 16-31
#pragma unroll
        for (int v = 0; v < 8; ++v) {
            const int m = (lane < 16) ? v : (v + 8);
            zx[(size_t)(row0 + m) * ZSTRIDE + col] = acc[v] + bias;
        }
    }
}

// ---------------------------------------------------------------------------
// Phase 2: sequential scan. One batch element per lane; 16 single-wave blocks.
// z-row loads software-pipelined one step ahead of the recurrence compute.
// ---------------------------------------------------------------------------
__device__ __forceinline__ float fast_rcp(float x) {
    return __builtin_amdgcn_rcpf(x);
}
__device__ __forceinline__ float sigm(float x) {
    return fast_rcp(1.0f + __expf(-x));
}
__device__ __forceinline__ float tanh_fast(float x) {
    const float e = __expf(2.0f * x);
    return 1.0f - 2.0f * fast_rcp(e + 1.0f);
}

__global__ __launch_bounds__(32) void qlstm_recur_kernel(
    const float* __restrict__ zx,
    const float* __restrict__ Wf, const float* __restrict__ Wi,
    const float* __restrict__ Wu, const float* __restrict__ Wo,
    const float* __restrict__ qwf, const float* __restrict__ qwi,
    const float* __restrict__ qwo,
    float* __restrict__ out)   // outputs [T*B*8] ++ hx [B*8] ++ cx [B*8]
{
    const int b = blockIdx.x * 32 + threadIdx.x;

    // Recurrent weights (wave-uniform -> scalar regs). v2f pairs encourage
    // v_pk_fma_f32 on the 8x8 matvec.
    v2f w2[4][NH];                       // w2[p][j] = {Whu[2p][j], Whu[2p+1][j]}
#pragma unroll
    for (int p = 0; p < 4; ++p)
#pragma unroll
        for (int j = 0; j < NH; ++j) {
            w2[p][j].x = Wu[(2 * p)     * FAN + DIM + j];
            w2[p][j].y = Wu[(2 * p + 1) * FAN + DIM + j];
        }
    float whf[NH], whi[NH], who[NH];
#pragma unroll
    for (int j = 0; j < NH; ++j) {
        whf[j] = Wf[DIM + j];
        whi[j] = Wi[DIM + j];
        who[j] = Wo[DIM + j];
    }
    const float cqf = __cosf(qwf[0]);
    const float cqi = __cosf(qwi[0]);
    const float cqo = __cosf(qwo[0]);

    float hx[NH], cx[NH];
#pragma unroll
    for (int h = 0; h < NH; ++h) { hx[h] = 0.0f; cx[h] = 0.0f; }

    const float* zp = zx + (size_t)b * ZSTRIDE;
    v4f za = *(const v4f*)(zp);
    v4f zb = *(const v4f*)(zp + 4);
    v4f zc = *(const v4f*)(zp + 8);

    for (int t = 0; t < T_STEPS; ++t) {
        // Issue next-step loads before this step's ~150-op compute
        const float* zn = zp + (size_t)BATCH * ZSTRIDE;
        v4f na = za, nb = zb, nc = zc;
        if (t + 1 < T_STEPS) {
            __builtin_prefetch(zn + BATCH * ZSTRIDE, 0, 3);  // t+2 into cache
            na = *(const v4f*)(zn);
            nb = *(const v4f*)(zn + 4);
            nc = *(const v4f*)(zn + 8);
        }

        v2f zg2[4] = { {za.x, za.y}, {za.z, za.w}, {zb.x, zb.y}, {zb.z, zb.w} };
        float zf = zc.x, zi = zc.y, zo = zc.z;

#pragma unroll
        for (int j = 0; j < NH; ++j) {
            const float hj = hx[j];
            zf = __builtin_fmaf(whf[j], hj, zf);
            zi = __builtin_fmaf(whi[j], hj, zi);
            zo = __builtin_fmaf(who[j], hj, zo);
            v2f hj2; hj2.x = hj; hj2.y = hj;
#pragma unroll
            for (int p = 0; p < 4; ++p)
                zg2[p] = zg2[p] + w2[p][j] * hj2;   // -> v_pk_fma_f32
        }

        const float f = sigm(__cosf(zf) * cqf);
        const float i = sigm(__cosf(zi) * cqi);
        const float o = sigm(__cosf(zo) * cqo);

        float zg[NH] = { zg2[0].x, zg2[0].y, zg2[1].x, zg2[1].y,
                         zg2[2].x, zg2[2].y, zg2[3].x, zg2[3].y };
#pragma unroll
        for (int h = 0; h < NH; ++h) {
            const float g = tanh_fast(zg[h]);
            const float c = __builtin_fmaf(f, cx[h], i * g);
            cx[h] = c;
            hx[h] = o * tanh_fast(c);
        }

        float* op = out + ((size_t)t * BATCH + b) * NH;
        v4f h0 = { hx[0], hx[1], hx[2], hx[3] };
        v4f h1 = { hx[4], hx[5], hx[6], hx[7] };
        *(v4f*)(op)     = h0;     // consecutive b -> contiguous 32B, coalesced
        *(v4f*)(op + 4) = h1;

        za = na; zb = nb; zc = nc;
        zp = zn;
    }

    // Final states: d_out = outputs ++ hx ++ cx
    float* hp = out + (size_t)T_STEPS * BATCH * NH + (size_t)b * NH;
    float* cp = hp + (size_t)BATCH * NH;
    v4f h0 = { hx[0], hx[1], hx[2], hx[3] };
    v4f h1 = { hx[4], hx[5], hx[6], hx[7] };
    v4f c0 = { cx[0], cx[1], cx[2], cx[3] };
    v4f c1 = { cx[4], cx[5], cx[6], cx[7] };
    *(v4f*)(hp)     = h0;  *(v4f*)(hp + 4) = h1;
    *(v4f*)(cp)     = c0;  *(v4f*)(cp + 4) = c1;
}

// ---------------------------------------------------------------------------
extern "C" void kernel_launch(void* const* d_in, const int* in_sizes, int n_in,
                              void* d_out, int out_size, void* d_ws, size_t ws_size,
                              hipStream_t stream) {
    const float* x   = (const float*)d_in[0];
    const float* Wf  = (const float*)d_in[1];
    const float* bf  = (const float*)d_in[2];
    const float* Wi  = (const float*)d_in[3];
    const float* bi  = (const float*)d_in[4];
    const float* Wu  = (const float*)d_in[5];
    const float* bu  = (const float*)d_in[6];
    const float* Wo  = (const float*)d_in[7];
    const float* bo  = (const float*)d_in[8];
    const float* qwf = (const float*)d_in[9];
    const float* qwi = (const float*)d_in[10];
    // d_in[11] = qwu (unused by the math: g has no quantum gate)
    const float* qwo = (const float*)d_in[12];

    float* out = (float*)d_out;
    float* zx  = (float*)d_ws;   // needs T*B*16*4 = 32 MiB

    // Phase 1: 32768 tiles / (8 waves * 4 tiles per wave) = 1024 blocks
    qlstm_gemm_kernel<<<1024, 256, 0, stream>>>(
        x, Wf, bf, Wi, bi, Wu, bu, Wo, bo, zx);

    // Phase 2: 512 batch lanes, one wave per block spread over WGPs
    qlstm_recur_kernel<<<16, 32, 0, stream>>>(
        zx, Wf, Wi, Wu, Wo, qwf, qwi, qwo, out);
}